// Discriminator_79113297592748
// MI455X (gfx1250) — compile-verified
//
#include <hip/hip_runtime.h>
#include <hip/hip_bf16.h>

// ---- shapes from the reference ----
#define B_ROWS  1024
#define IN_DIM  165
#define EMBD    150
#define FEAT    315      // 165 + 150
#define H0_COLS 328      // 325 (FEAT + MB_OUT) padded to a multiple of 8
#define MB_OUT  10
#define MB_INT  50
#define MB_N    500      // MB_OUT * MB_INT (logical)
#define MB_LD   512      // padded leading dim of the M buffer

typedef __attribute__((ext_vector_type(2))) float v2f;
typedef __attribute__((ext_vector_type(8))) float v8f;

// ------------------------------------------------------------------
// 1) embedding lookup + concat + zero-pad into X[1024][328]
//    cols 0..164 = time_series, 165..314 = emb[cond], 315..327 = 0
//    (cols 315..324 are later overwritten with minibatch features)
// ------------------------------------------------------------------
__global__ void embed_concat_kernel(const float* __restrict__ ts,
                                    const int*   __restrict__ cond,
                                    const float* __restrict__ emb,
                                    float* __restrict__ X) {
  int idx = blockIdx.x * blockDim.x + threadIdx.x;
  if (idx >= B_ROWS * H0_COLS) return;
  int i = idx / H0_COLS;
  int f = idx - i * H0_COLS;
  float v = 0.0f;
  if (f < IN_DIM)      v = ts[i * IN_DIM + f];
  else if (f < FEAT)   v = emb[cond[i] * EMBD + (f - IN_DIM)];
  X[idx] = v;
}

// ------------------------------------------------------------------
// 2) Repack B[KxN] (row-major) into zero-padded K-pair-interleaved
//    Bp[Kp/2][Np][2]:  Bp[p][col] = (B[2p][col], B[2p+1][col])
//    so a WMMA B-fragment load is a single aligned b64.
// ------------------------------------------------------------------
__global__ void pack_pairs_kernel(const float* __restrict__ B, int K, int N,
                                  float* __restrict__ Bp, int Kp, int Np) {
  int idx = blockIdx.x * blockDim.x + threadIdx.x;
  int total = (Kp >> 1) * Np;
  if (idx >= total) return;
  int p   = idx / Np;
  int col = idx - p * Np;
  int k0 = 2 * p;
  v2f v;
  v.x = (k0     < K && col < N) ? B[k0       * N + col] : 0.f;
  v.y = (k0 + 1 < K && col < N) ? B[(k0 + 1) * N + col] : 0.f;
  *(v2f*)(Bp + (size_t)idx * 2) = v;
}

// ------------------------------------------------------------------
// 3) f32 WMMA GEMM on pre-padded operands:
//      C = act(A[1024 x Kp] * B[Kp x Np] + bias), store only cols < N
//    One wave = one 16x16 tile (V_WMMA_F32_16X16X4_F32), 8 waves/block.
//    Inner loop: 2x global_load_b64 + 1x wmma, branch-free.
//    act: 0 = none, 1 = leaky_relu(0.2), 2 = sigmoid
// ------------------------------------------------------------------
__global__ void wmma_gemm_packed(const float* __restrict__ A, int lda,
                                 const float* __restrict__ Bp, int Np,
                                 const float* __restrict__ bias,
                                 float* __restrict__ C, int ldc,
                                 int N, int Kp, int act) {
  const int lane = threadIdx.x & 31;
  const int wave = threadIdx.x >> 5;
  const int half = lane >> 4;     // 16-lane half: A lanes0-15 K={k,k+1}, lanes16-31 K={k+2,k+3}
  const int l15  = lane & 15;
  const int tm = blockIdx.y * 16;              // M = 1024, always exact
  const int tn = blockIdx.x * 128 + wave * 16;
  if (tn >= Np) return;                        // wave-uniform: EXEC all-1s at WMMA

  const int row = tm + l15;
  const int col = tn + l15;

  // branch-free, 8B-aligned fragment pointers
  const float* ap = A  + (size_t)row * lda + 2 * half;           // A[row][k+2h .. +1]
  const float* bp = Bp + ((size_t)half * Np + col) * 2;          // pair p = k/2 + h, col
  const size_t bstep = 4 * (size_t)Np;                           // p advances by 2 per k+=4

  v8f acc = {};
#pragma unroll 4
  for (int k = 0; k < Kp; k += 4) {
    v2f a = *(const v2f*)ap;
    v2f b = *(const v2f*)bp;
    ap += 4;
    bp += bstep;
    acc = __builtin_amdgcn_wmma_f32_16x16x4_f32(
        /*neg_a=*/false, a, /*neg_b=*/false, b,
        /*c_mod=*/(short)0, acc, /*reuse_a=*/false, /*reuse_b=*/false);
  }

  if (col >= N) return;
  const float bv = bias ? bias[col] : 0.f;
#pragma unroll
  for (int r = 0; r < 8; ++r) {
    // C/D layout: VGPR r, lanes 0-15 -> M=tm+r, lanes 16-31 -> M=tm+r+8
    float v = acc[r] + bv;
    if (act == 1)      v = (v > 0.f) ? v : 0.2f * v;
    else if (act == 2) v = 1.0f / (1.0f + __expf(-v));
    C[(size_t)(tm + r + 8 * half) * ldc + col] = v;
  }
}

// ------------------------------------------------------------------
// 4) Minibatch discrimination pairwise stage.
//    Block = (128-row i-tile, one output channel o).
//    mb[i][o] = sum_j exp( -sum_k |M[i][o*50+k] - M[j][o*50+k]| )
//    j-tiles staged through LDS; result written into X[:,315+o].
// ------------------------------------------------------------------
__global__ void pairwise_kernel(const float* __restrict__ Mbuf,
                                float* __restrict__ X) {
  __shared__ float Mj[128][MB_INT + 2];   // padded stride vs bank conflicts
  const int o = blockIdx.y;
  const int i = blockIdx.x * 128 + threadIdx.x;

  float mi[MB_INT];
#pragma unroll
  for (int k = 0; k < MB_INT; ++k)
    mi[k] = Mbuf[(size_t)i * MB_LD + o * MB_INT + k];

  float accsum = 0.f;
  for (int j0 = 0; j0 < B_ROWS; j0 += 128) {
    __syncthreads();
    for (int t = threadIdx.x; t < 128 * MB_INT; t += 128) {
      const int jr = t / MB_INT;
      const int kc = t - jr * MB_INT;
      Mj[jr][kc] = Mbuf[(size_t)(j0 + jr) * MB_LD + o * MB_INT + kc];
    }
    __syncthreads();
#pragma unroll 2
    for (int j = 0; j < 128; ++j) {
      float s = 0.f;
#pragma unroll
      for (int k = 0; k < MB_INT; ++k)
        s += fabsf(mi[k] - Mj[j][k]);
      accsum += __expf(-s);
    }
  }
  // includes the self term exp(0)=1, matching the reference
  X[(size_t)i * H0_COLS + FEAT + o] = accsum;
}

// ------------------------------------------------------------------
// launcher — all on `stream`, deterministic, no allocation
// ------------------------------------------------------------------
static inline void run_pack(const float* B, int K, int N, float* Bp, int Kp, int Np,
                            hipStream_t stream) {
  int total = (Kp >> 1) * Np;
  pack_pairs_kernel<<<(total + 255) / 256, 256, 0, stream>>>(B, K, N, Bp, Kp, Np);
}

extern "C" void kernel_launch(void* const* d_in, const int* in_sizes, int n_in,
                              void* d_out, int out_size, void* d_ws, size_t ws_size,
                              hipStream_t stream) {
  (void)in_sizes; (void)n_in; (void)out_size; (void)ws_size;

  const float* ts   = (const float*)d_in[0];
  const int*   cond = (const int*)  d_in[1];
  const float* emb  = (const float*)d_in[2];
  const float* T    = (const float*)d_in[3];   // [315][10][50] == row-major [315][500]
  const float* W1   = (const float*)d_in[4];
  const float* b1   = (const float*)d_in[5];
  const float* W2   = (const float*)d_in[6];
  const float* b2   = (const float*)d_in[7];
  const float* W3   = (const float*)d_in[8];
  const float* b3   = (const float*)d_in[9];
  const float* W4   = (const float*)d_in[10];
  const float* b4   = (const float*)d_in[11];
  const float* W5   = (const float*)d_in[12];
  const float* b5   = (const float*)d_in[13];
  float* out = (float*)d_out;

  // workspace layout (f32), ~23 MB
  float* X    = (float*)d_ws;                  // [1024][328]
  float* Mb   = X    + B_ROWS * H0_COLS;       // [1024][512]
  float* h1   = Mb   + B_ROWS * MB_LD;         // [1024][1536]
  float* h2   = h1   + B_ROWS * 1536;          // [1024][1024]
  float* h3   = h2   + B_ROWS * 1024;          // [1024][512]
  float* h4   = h3   + B_ROWS * 512;           // [1024][256]
  float* Wpad = h4   + B_ROWS * 256;           // [1536*1024] shared pad/pack buffer

  const dim3 blk(256);

  embed_concat_kernel<<<(B_ROWS * H0_COLS + 255) / 256, 256, 0, stream>>>(ts, cond, emb, X);

  // M = X @ T   (K 315->320, N 500->512)
  run_pack(T, FEAT, MB_N, Wpad, 320, MB_LD, stream);
  wmma_gemm_packed<<<dim3(4, 64), blk, 0, stream>>>(X, H0_COLS, Wpad, MB_LD,
                                                    nullptr, Mb, MB_LD, MB_N, 320, 0);

  // minibatch features -> X cols 315..324
  pairwise_kernel<<<dim3(B_ROWS / 128, MB_OUT), 128, 0, stream>>>(Mb, X);

  // h1 = lrelu(X @ W1 + b1)    K 325->328, N 1536
  run_pack(W1, 325, 1536, Wpad, H0_COLS, 1536, stream);
  wmma_gemm_packed<<<dim3(12, 64), blk, 0, stream>>>(X, H0_COLS, Wpad, 1536, b1, h1, 1536, 1536, H0_COLS, 1);

  // h2 = lrelu(h1 @ W2 + b2)   K 1536, N 1024
  run_pack(W2, 1536, 1024, Wpad, 1536, 1024, stream);
  wmma_gemm_packed<<<dim3(8, 64), blk, 0, stream>>>(h1, 1536, Wpad, 1024, b2, h2, 1024, 1024, 1536, 1);

  // h3 = lrelu(h2 @ W3 + b3)   K 1024, N 512
  run_pack(W3, 1024, 512, Wpad, 1024, 512, stream);
  wmma_gemm_packed<<<dim3(4, 64), blk, 0, stream>>>(h2, 1024, Wpad, 512, b3, h3, 512, 512, 1024, 1);

  // h4 = lrelu(h3 @ W4 + b4)   K 512, N 256
  run_pack(W4, 512, 256, Wpad, 512, 256, stream);
  wmma_gemm_packed<<<dim3(2, 64), blk, 0, stream>>>(h3, 512, Wpad, 256, b4, h4, 256, 256, 512, 1);

  // out = sigmoid(h4 @ W5 + b5)  K 256, N 1->16
  run_pack(W5, 256, 1, Wpad, 256, 16, stream);
  wmma_gemm_packed<<<dim3(1, 64), blk, 0, stream>>>(h4, 256, Wpad, 16, b5, out, 1, 1, 256, 2);
}